// MaskedBatchNorm2d_25228637896861
// MI455X (gfx1250) — compile-verified
//
#include <hip/hip_runtime.h>
#include <stdint.h>

// MaskedBatchNorm2d for x[8,64,496,432] f32 on MI455X (gfx1250).
//
// Bandwidth-bound: 439 MB/pass. One fused stats pass (sum & sum-of-squares ->
// var = S2/c - mean^2, pad terms added analytically from the 512 x00 values),
// one normalize pass: ~1.29 GB total -> ~55us at 23.3 TB/s.
// CDNA5 path used: Tensor Data Mover (tensor_load_to_lds) with the LDS pad
// feature (row stride 257 dwords) so both LDS read phases are bank-conflict
// free, plus s_wait_tensorcnt for async pipelining. No WMMA on purpose: the op
// has no matmul structure and is HBM-bound.
// Normalize pass streams float2 (b64, 256B/wave/instr) to halve VMEM issue.

#define CH 64
#define BATCH 8
#define PLANE 214272          // 496*432 contiguous floats per (b,c) plane
#define TILE 256              // spatial positions per TDM tile
#define ROWPAD 257            // LDS row stride in dwords (TDM pads +1 dword / 256)
#define TILES_PER_BATCH 837   // 214272 / 256
#define BLOCKS_X 27
#define TILES_PER_BLOCK 31    // 27*31 = 837
#define NBLK (BLOCKS_X * BATCH)  // 216
#define NORM_BLOCK 192
#define NORM_GRID_X 558       // 558*192*2 = 214272

typedef __attribute__((ext_vector_type(4))) unsigned int u32x4;
typedef __attribute__((ext_vector_type(8))) int          i32x8;
typedef __attribute__((ext_vector_type(4))) int          i32x4;

// Issue one TDM 2-D tile load: 64 rows (channels, stride PLANE floats) x
// 256 contiguous floats -> LDS at ldsByteOff, padded to 257-dword rows.
__device__ __forceinline__ void tdm_load_tile(const float* gbase, unsigned ldsByteOff)
{
#if defined(__HIP_DEVICE_COMPILE__)
  uint64_t ga = (uint64_t)(uintptr_t)gbase;
  u32x4 g0;
  g0[0] = 1u;                                            // count=1, user mode, no gather
  g0[1] = ldsByteOff;                                    // lds_addr (bytes)
  g0[2] = (unsigned)(ga & 0xFFFFFFFFu);                  // global_addr[31:0]
  g0[3] = (unsigned)((ga >> 32) & 0x01FFFFFFu)           // global_addr[56:32]
        | (2u << 30);                                    // type = 2 ("image")
  i32x8 g1;
  g1[0] = (int)((2u << 16)      // data_size = 4B
              | (1u << 20)      // pad_enable
              | (7u << 22));    // pad_interval = 256 dwords (pad_amount=0 -> 1 dword)
  g1[1] = (int)((PLANE & 0xFFFF) << 16);                 // tensor_dim0[15:0]
  g1[2] = (int)((PLANE >> 16) | (CH << 16));             // tensor_dim0[31:16] | tensor_dim1
  g1[3] = (int)(TILE << 16);                             // tile_dim0 = 256
  g1[4] = (int)CH;                                       // tile_dim1 = 64, tile_dim2 = 0
  g1[5] = (int)PLANE;                                    // tensor_dim0_stride[31:0]
  g1[6] = 0;                                             // stride hi / dim1_stride lo
  g1[7] = 0;
  i32x4 z4 = {0, 0, 0, 0};
#if __clang_major__ >= 23
  i32x8 z8 = {0, 0, 0, 0, 0, 0, 0, 0};
  __builtin_amdgcn_tensor_load_to_lds(g0, g1, z4, z4, z8, 0);
#else
  __builtin_amdgcn_tensor_load_to_lds(g0, g1, z4, z4, 0);
#endif
#endif
}

__device__ __forceinline__ void wait_tensorcnt0()
{
#if defined(__HIP_DEVICE_COMPILE__)
  __builtin_amdgcn_s_wait_tensorcnt(0);
#endif
}

// ---------------------------------------------------------------------------
// Kernel 1: per-block partial stats. grid = (27, 8), block = 256.
// Dynamic LDS: 2 tile buffers (64*257 floats each) + mask[256] + red1[256] +
// red2[256] + cnt[256] = 33920 floats = 135,680 B (<160 KB: 2 blocks/WGP).
// ---------------------------------------------------------------------------
__global__ void __launch_bounds__(256)
mbn_stats_kernel(const float* __restrict__ x, float* __restrict__ p1,
                 float* __restrict__ p2, unsigned* __restrict__ cnts)
{
  extern __shared__ float smem[];
  float*    buf0    = smem;
  float*    buf1    = smem + CH * ROWPAD;
  float*    maskArr = smem + 2 * CH * ROWPAD;   // 256 floats
  float*    red1    = maskArr + TILE;           // 256 floats
  float*    red2    = red1 + 256;               // 256 floats
  unsigned* cntArr  = (unsigned*)(red2 + 256);  // 256 u32

  const int tid   = threadIdx.x;
  const int b     = blockIdx.y;
  const int tile0 = blockIdx.x * TILES_PER_BLOCK;
  const float* gb = x + (uint64_t)b * CH * PLANE;

  // SGPR-uniform wave-0 test: TDM ignores EXEC, so only ONE wave may reach
  // the tensor_load_to_lds instruction (scalar branch guaranteed).
  const bool isWave0 = ((__builtin_amdgcn_readfirstlane((int)threadIdx.x) >> 5) == 0);

  const int c = tid & (CH - 1);   // channel owned in phase B
  const int g = tid >> 6;         // quarter of the tile owned in phase B
  float a1 = 0.0f, a2 = 0.0f;
  unsigned localCnt = 0u;

  if (isWave0) {
    tdm_load_tile(gb + (uint64_t)tile0 * TILE, (unsigned)(uintptr_t)buf0);
  }

  for (int t = 0; t < TILES_PER_BLOCK; ++t) {
    const int cur = t & 1;
    float* tb = cur ? buf1 : buf0;
    float* nb = cur ? buf0 : buf1;

    if (isWave0) wait_tensorcnt0();   // tile t landed in LDS
    __syncthreads();                  // release all waves; nb free of readers
    asm volatile("" ::: "memory");    // LDS written behind the compiler's back

    if (isWave0 && (t + 1) < TILES_PER_BLOCK) {   // prefetch tile t+1
      tdm_load_tile(gb + (uint64_t)(tile0 + t + 1) * TILE,
                    (unsigned)(uintptr_t)nb);
    }

    // Phase A: thread = position. Banks (c + tid) % 64 -> conflict-free.
    float s = 0.0f;
    #pragma unroll
    for (int cc = 0; cc < CH; ++cc) s += tb[cc * ROWPAD + tid];
    const float m = (s != 0.0f) ? 1.0f : 0.0f;
    maskArr[tid] = m;
    localCnt += (unsigned)m;
    __syncthreads();

    // Phase B: thread = (channel c, quarter g). Banks (c + i) % 64 -> clean.
    const float* row  = tb + c * ROWPAD + g * 64;
    const float* mrow = maskArr + g * 64;
    #pragma unroll
    for (int i = 0; i < 64; ++i) {
      const float v  = row[i];
      const float mm = mrow[i];
      a1 = fmaf(mm, v, a1);
      a2 = fmaf(mm * v, v, a2);
    }
    // no trailing barrier: next iteration's top barrier orders reuse
  }

  red1[c * 4 + g] = a1;
  red2[c * 4 + g] = a2;
  cntArr[tid] = localCnt;
  __syncthreads();

  const int blkId = b * BLOCKS_X + blockIdx.x;
  if (tid < CH) {
    p1[blkId * CH + tid] = red1[tid * 4] + red1[tid * 4 + 1] +
                           red1[tid * 4 + 2] + red1[tid * 4 + 3];
    p2[blkId * CH + tid] = red2[tid * 4] + red2[tid * 4 + 1] +
                           red2[tid * 4 + 2] + red2[tid * 4 + 3];
  }
  if (tid == 0) {
    unsigned n = 0;
    for (int i = 0; i < 256; ++i) n += cntArr[i];
    cnts[blkId] = n;
  }
}

// ---------------------------------------------------------------------------
// Kernel 2: finalize. 1 block, 64 threads. Deterministic fixed-order reduce.
// ---------------------------------------------------------------------------
__global__ void __launch_bounds__(64)
mbn_finalize_kernel(const float* __restrict__ x, const float* __restrict__ p1,
                    const float* __restrict__ p2, const unsigned* __restrict__ cnts,
                    float* __restrict__ invOut, float* __restrict__ flagOut)
{
  __shared__ float nsh[BATCH];
  __shared__ float padsh[BATCH];
  const int t = threadIdx.x;

  if (t < BATCH) {
    unsigned n = 0;
    for (int i = 0; i < BLOCKS_X; ++i) n += cnts[t * BLOCKS_X + i];
    nsh[t] = (float)n;
  }
  __syncthreads();

  float N = nsh[0];
  for (int i = 1; i < BATCH; ++i) N = fmaxf(N, nsh[i]);

  if (t < BATCH) {
    const float pad = N - nsh[t];
    padsh[t] = pad;
    float s00 = 0.0f;                      // mask at (0,0): channel sum of x00
    for (int cc = 0; cc < CH; ++cc) s00 += x[((uint64_t)t * CH + cc) * PLANE];
    flagOut[t] = ((pad > 0.0f) || (s00 != 0.0f)) ? 1.0f : 0.0f;
  }
  __syncthreads();

  // thread t = channel: S1, S2 over all 216 blocks (fixed order) + pad terms
  float S1 = 0.0f, S2 = 0.0f;
  for (int k = 0; k < NBLK; ++k) { S1 += p1[k * CH + t]; S2 += p2[k * CH + t]; }
  for (int b = 0; b < BATCH; ++b) {
    const float x00 = x[((uint64_t)b * CH + t) * PLANE];
    S1 += padsh[b] * x00;
    S2 += padsh[b] * x00 * x00;
  }
  const float count = (float)BATCH * N;
  const float mean  = S1 / count;
  float var = S2 / count - mean * mean;
  var = fmaxf(var, 0.0f);
  invOut[t] = rsqrtf(var + 0.001f);
}

// ---------------------------------------------------------------------------
// Kernel 3: normalize. grid = (558, 8), block = 192. Register streaming with
// float2 (b64): each thread owns 2 consecutive spatial positions for all 64
// channels -> 256B per wave per VMEM instruction, half the issue count of b32.
// 558*192*2 = 214272 positions exactly, no tail.
// ---------------------------------------------------------------------------
__global__ void __launch_bounds__(NORM_BLOCK)
mbn_norm_kernel(const float* __restrict__ x, float* __restrict__ out,
                const float* __restrict__ inv, const float* __restrict__ flags)
{
  __shared__ float sInv[CH];
  __shared__ float sFlag;
  const int b = blockIdx.y;
  const long p = (long)blockIdx.x * NORM_BLOCK + threadIdx.x;  // pair index

  if (threadIdx.x < CH) sInv[threadIdx.x] = inv[threadIdx.x];
  if (threadIdx.x == 0) sFlag = flags[b];
  __syncthreads();

  const float2* base = (const float2*)(x + (uint64_t)b * CH * PLANE) + p;
  float2 v[CH];
  float s0 = 0.0f, s1 = 0.0f;
  #pragma unroll
  for (int cc = 0; cc < CH; ++cc) {
    v[cc] = base[(uint64_t)cc * (PLANE / 2)];
    s0 += v[cc].x;
    s1 += v[cc].y;
  }
  // mask, plus the (0,0) override: pad>0 or mask00 forces x00*inv at pos 0
  const bool sc0 = (s0 != 0.0f) || ((p == 0) && (sFlag != 0.0f));
  const bool sc1 = (s1 != 0.0f);

  float2* ob = (float2*)(out + (uint64_t)b * CH * PLANE) + p;
  #pragma unroll
  for (int cc = 0; cc < CH; ++cc) {
    float2 r;
    r.x = sc0 ? v[cc].x * sInv[cc] : v[cc].x;
    r.y = sc1 ? v[cc].y * sInv[cc] : v[cc].y;
    ob[(uint64_t)cc * (PLANE / 2)] = r;
  }
}

// ---------------------------------------------------------------------------
extern "C" void kernel_launch(void* const* d_in, const int* in_sizes, int n_in,
                              void* d_out, int out_size, void* d_ws, size_t ws_size,
                              hipStream_t stream)
{
  const float* x  = (const float*)d_in[0];
  float* out      = (float*)d_out;

  // workspace layout (all regions fully rewritten every call): ~112 KB
  float*    p1    = (float*)d_ws;            // 216*64
  float*    p2    = p1 + NBLK * CH;          // 216*64
  unsigned* cnts  = (unsigned*)(p2 + NBLK * CH);  // 216
  float*    inv   = (float*)(cnts + NBLK);   // 64
  float*    flags = inv + CH;                // 8

  const size_t smemBytes = (size_t)(2 * CH * ROWPAD + 4 * 256) * sizeof(float); // 135,680 B

  dim3 g1(BLOCKS_X, BATCH), b1(256);
  mbn_stats_kernel<<<g1, b1, smemBytes, stream>>>(x, p1, p2, cnts);

  mbn_finalize_kernel<<<dim3(1), dim3(64), 0, stream>>>(x, p1, p2, cnts, inv, flags);

  dim3 g3(NORM_GRID_X, BATCH), b3(NORM_BLOCK);
  mbn_norm_kernel<<<g3, b3, 0, stream>>>(x, out, inv, flags);
}